// GCNRouting2Hop_32375463477769
// MI455X (gfx1250) — compile-verified
//
#include <hip/hip_runtime.h>
#include <hip/hip_bf16.h>

typedef __attribute__((ext_vector_type(2))) float v2f;
typedef __attribute__((ext_vector_type(8))) float v8f;

#define DH 256

// ---------------- degree / normalization ----------------

__global__ void __launch_bounds__(256) k_deg_init(float* deg, int n) {
    int i = blockIdx.x * 256 + threadIdx.x;
    if (i < n) deg[i] = 1.0f;  // self-loop weight
}

__global__ void __launch_bounds__(256) k_deg_scatter(const long long* __restrict__ dst,
                                                     const float* __restrict__ w,
                                                     float* __restrict__ deg, int E) {
    int e = blockIdx.x * 256 + threadIdx.x;
    if (e < E) atomicAdd(&deg[dst[e]], w[e]);
}

__global__ void __launch_bounds__(256) k_dinv(float* deg, int n) {
    int i = blockIdx.x * 256 + threadIdx.x;
    if (i < n) {
        float d = deg[i];
        deg[i] = (d > 0.0f) ? rsqrtf(d) : 0.0f;
    }
}

// ---------------- WMMA GEMMs (fp32, 16x16x4) ----------------
// One wave per 16x16 output tile. Grid sized exactly -> EXEC all 1s.
// A [rows, K] row-major, W [K, 256] row-major, C [rows, 256] row-major.

// Dual-output GEMM: C1 = A@W1, C2 = A@W2 (shares A fragments). K = 128.
__global__ void __launch_bounds__(128) k_gemm_dual_k128(const float* __restrict__ A,
                                                        const float* __restrict__ W1,
                                                        const float* __restrict__ W2,
                                                        float* __restrict__ C1,
                                                        float* __restrict__ C2) {
    const int wave = blockIdx.x * 4 + (threadIdx.x >> 5);
    const int lane = threadIdx.x & 31;
    const int tm = wave >> 4;        // 16 N-tiles (256/16)
    const int tn = wave & 15;
    const int half = lane >> 4;      // 0: lanes 0-15, 1: lanes 16-31
    const int l16 = lane & 15;
    const int m = tm * 16 + l16;
    const int n = tn * 16 + l16;
    const int K = 128;

    v8f c1 = {};
    v8f c2 = {};
    const float* arow = A + (size_t)m * K;

    for (int k0 = 0; k0 < K; k0 += 4) {
        const int kb = k0 + half * 2;           // A: VGPR0/1 = K=kb, kb+1
        v2f a = *(const v2f*)(arow + kb);       // 8B-aligned (kb even)
        v2f b1, b2;
        b1.x = W1[(size_t)kb * DH + n];
        b1.y = W1[(size_t)(kb + 1) * DH + n];
        b2.x = W2[(size_t)kb * DH + n];
        b2.y = W2[(size_t)(kb + 1) * DH + n];
        c1 = __builtin_amdgcn_wmma_f32_16x16x4_f32(false, a, false, b1, (short)0, c1,
                                                   false, false);
        c2 = __builtin_amdgcn_wmma_f32_16x16x4_f32(false, a, false, b2, (short)0, c2,
                                                   false, false);
    }

    const int rbase = tm * 16 + half * 8;       // C/D: VGPR r -> M = r + 8*half
#pragma unroll
    for (int r = 0; r < 8; ++r) {
        C1[(size_t)(rbase + r) * DH + n] = c1[r];
        C2[(size_t)(rbase + r) * DH + n] = c2[r];
    }
}

// Single-output GEMM, K = 256 (h @ W2).
__global__ void __launch_bounds__(128) k_gemm_k256(const float* __restrict__ A,
                                                   const float* __restrict__ W,
                                                   float* __restrict__ C) {
    const int wave = blockIdx.x * 4 + (threadIdx.x >> 5);
    const int lane = threadIdx.x & 31;
    const int tm = wave >> 4;
    const int tn = wave & 15;
    const int half = lane >> 4;
    const int l16 = lane & 15;
    const int m = tm * 16 + l16;
    const int n = tn * 16 + l16;
    const int K = 256;

    v8f c = {};
    const float* arow = A + (size_t)m * K;

    for (int k0 = 0; k0 < K; k0 += 4) {
        const int kb = k0 + half * 2;
        v2f a = *(const v2f*)(arow + kb);
        v2f b;
        b.x = W[(size_t)kb * DH + n];
        b.y = W[(size_t)(kb + 1) * DH + n];
        c = __builtin_amdgcn_wmma_f32_16x16x4_f32(false, a, false, b, (short)0, c,
                                                  false, false);
    }

    const int rbase = tm * 16 + half * 8;
#pragma unroll
    for (int r = 0; r < 8; ++r) {
        C[(size_t)(rbase + r) * DH + n] = c[r];
    }
}

// ---------------- edge scatter (sym-normalized aggregation) ----------------
// One block per (edge or self-loop); thread t owns feature dim t.
// agg[dst, :] += dinv[src]*w*dinv[dst] * h[src, :]

__global__ void __launch_bounds__(256) k_scatter(const long long* __restrict__ src,
                                                 const long long* __restrict__ dst,
                                                 const float* __restrict__ w,
                                                 const float* __restrict__ dinv,
                                                 const float* __restrict__ h,
                                                 float* __restrict__ agg, int E) {
    const int e = blockIdx.x;
    const int t = threadIdx.x;
    long long s, d;
    float wt;
    if (e < E) {
        s = src[e];
        d = dst[e];
        wt = w[e];
    } else {           // self loop, weight 1
        s = d = (long long)(e - E);
        wt = 1.0f;
    }
    const float norm = dinv[s] * wt * dinv[d];
    const float val = norm * h[(size_t)s * DH + t];
    atomicAdd(&agg[(size_t)d * DH + t], val);
}

// ---------------- fused epilogues: bias (+ReLU) + residual + LayerNorm ----
// One wave (32 lanes) per node; 8 dims per lane. wave32 shuffle reductions.

__device__ __forceinline__ float wave_sum32(float v) {
#pragma unroll
    for (int m = 16; m >= 1; m >>= 1) v += __shfl_xor(v, m, 32);
    return v;
}

// hop0: delta = relu(agg + b1); v = hres + bres + delta; out = LN(v)*g + beta
__global__ void __launch_bounds__(256) k_post0(const float* __restrict__ agg,
                                               const float* __restrict__ hres,
                                               const float* __restrict__ b1,
                                               const float* __restrict__ bres,
                                               const float* __restrict__ gamma,
                                               const float* __restrict__ beta,
                                               float* __restrict__ out, int n) {
    const int node = blockIdx.x * 8 + (threadIdx.x >> 5);
    const int lane = threadIdx.x & 31;
    if (node >= n) return;
    const size_t base = (size_t)node * DH;
    float v[8];
    float sum = 0.0f;
#pragma unroll
    for (int j = 0; j < 8; ++j) {
        const int idx = lane + j * 32;
        float delta = agg[base + idx] + b1[idx];
        delta = delta > 0.0f ? delta : 0.0f;
        v[j] = hres[base + idx] + bres[idx] + delta;
        sum += v[j];
    }
    const float mu = wave_sum32(sum) * (1.0f / DH);
    float var = 0.0f;
#pragma unroll
    for (int j = 0; j < 8; ++j) {
        const float dv = v[j] - mu;
        var += dv * dv;
    }
    var = wave_sum32(var) * (1.0f / DH);
    const float inv = rsqrtf(var + 1e-5f);
#pragma unroll
    for (int j = 0; j < 8; ++j) {
        const int idx = lane + j * 32;
        out[base + idx] = (v[j] - mu) * inv * gamma[idx] + beta[idx];
    }
}

// hop1: v = h + agg + b2; out = LN(v)*g + beta   (no relu)
__global__ void __launch_bounds__(256) k_post1(const float* __restrict__ agg,
                                               const float* __restrict__ h,
                                               const float* __restrict__ b2,
                                               const float* __restrict__ gamma,
                                               const float* __restrict__ beta,
                                               float* __restrict__ out, int n) {
    const int node = blockIdx.x * 8 + (threadIdx.x >> 5);
    const int lane = threadIdx.x & 31;
    if (node >= n) return;
    const size_t base = (size_t)node * DH;
    float v[8];
    float sum = 0.0f;
#pragma unroll
    for (int j = 0; j < 8; ++j) {
        const int idx = lane + j * 32;
        v[j] = h[base + idx] + agg[base + idx] + b2[idx];
        sum += v[j];
    }
    const float mu = wave_sum32(sum) * (1.0f / DH);
    float var = 0.0f;
#pragma unroll
    for (int j = 0; j < 8; ++j) {
        const float dv = v[j] - mu;
        var += dv * dv;
    }
    var = wave_sum32(var) * (1.0f / DH);
    const float inv = rsqrtf(var + 1e-5f);
#pragma unroll
    for (int j = 0; j < 8; ++j) {
        const int idx = lane + j * 32;
        out[base + idx] = (v[j] - mu) * inv * gamma[idx] + beta[idx];
    }
}

// ---------------- host launch ----------------

extern "C" void kernel_launch(void* const* d_in, const int* in_sizes, int n_in,
                              void* d_out, int out_size, void* d_ws, size_t ws_size,
                              hipStream_t stream) {
    const float* x       = (const float*)d_in[0];
    const long long* ei  = (const long long*)d_in[1];
    const float* ew      = (const float*)d_in[2];
    const float* W1      = (const float*)d_in[3];
    const float* b1      = (const float*)d_in[4];
    const float* W2      = (const float*)d_in[5];
    const float* b2      = (const float*)d_in[6];
    const float* Wres    = (const float*)d_in[7];
    const float* bres    = (const float*)d_in[8];
    const float* g1      = (const float*)d_in[9];
    const float* be1     = (const float*)d_in[10];
    const float* g2      = (const float*)d_in[11];
    const float* be2     = (const float*)d_in[12];

    const int N = in_sizes[0] / 128;   // 20000
    const int E = in_sizes[2];         // 320000
    const long long* src = ei;
    const long long* dst = ei + E;

    // workspace layout (fp32)
    float* ws = (float*)d_ws;
    const size_t pad = ((size_t)N + 255) / 256 * 256;
    const size_t ND = (size_t)N * DH;
    float* deg  = ws;              // N      (becomes dinv)
    float* h1   = ws + pad;        // N*256  (hop0 conv pre-agg; reused as h2)
    float* hres = h1 + ND;         // N*256  (x @ Wres)
    float* agg  = hres + ND;       // N*256  (scatter accumulator, both hops)
    float* h    = agg + ND;        // N*256  (hop0 LN output)
    float* out  = (float*)d_out;

    // 1) degrees -> dinv
    k_deg_init<<<(N + 255) / 256, 256, 0, stream>>>(deg, N);
    k_deg_scatter<<<(E + 255) / 256, 256, 0, stream>>>(dst, ew, deg, E);
    k_dinv<<<(N + 255) / 256, 256, 0, stream>>>(deg, N);

    // 2) h1 = x@W1, hres = x@Wres   (N/16 * 16 tiles = N waves, 4 waves/block)
    k_gemm_dual_k128<<<N / 4, 128, 0, stream>>>(x, W1, Wres, h1, hres);

    // 3) hop0 aggregation
    hipMemsetAsync(agg, 0, ND * sizeof(float), stream);
    k_scatter<<<E + N, 256, 0, stream>>>(src, dst, ew, deg, h1, agg, E);

    // 4) h = LN(x@Wres + bres + relu(agg + b1))
    k_post0<<<(N + 7) / 8, 256, 0, stream>>>(agg, hres, b1, bres, g1, be1, h, N);

    // 5) h2 = h @ W2  (reuse h1 buffer)
    k_gemm_k256<<<N / 4, 128, 0, stream>>>(h, W2, h1);

    // 6) hop1 aggregation
    hipMemsetAsync(agg, 0, ND * sizeof(float), stream);
    k_scatter<<<E + N, 256, 0, stream>>>(src, dst, ew, deg, h1, agg, E);

    // 7) out = LN(h + agg + b2)
    k_post1<<<(N + 7) / 8, 256, 0, stream>>>(agg, h, b2, g2, be2, out, N);
}